// TransformerBlock_64957085384896
// MI455X (gfx1250) — compile-verified
//
#include <hip/hip_runtime.h>
#include <math.h>

#define BB 2
#define TT 1024
#define EE 128
#define HH 4
#define HID 512
#define NK 4
#define VS 32

typedef float v2f __attribute__((ext_vector_type(2)));
typedef float v8f __attribute__((ext_vector_type(8)));

__device__ __forceinline__ unsigned pcg_hash(unsigned x) {
    x = x * 747796405u + 2891336453u;
    unsigned w = ((x >> ((x >> 28u) + 4u)) ^ x) * 277803737u;
    return (w >> 22u) ^ w;
}

__device__ __forceinline__ float softplusf(float x) {
    return (x > 20.0f) ? x : log1pf(__expf(x));
}

// ---------------------------------------------------------------------------
// fp32 WMMA GEMM, compile-time shape <N, KD>: C[M,N] = A[M,KD] @ W[KD,N].
// Block = 8 waves sharing one 16-column W panel staged in LDS via the CDNA5
// async global->LDS DMA path; each wave owns one 16x16 C tile (8 consecutive
// M tiles per block). Epilogue: (+coord*coordRow[n])*scale + bias, relu,
// +resid, optional head-major (b,h,t,e) store.
// ---------------------------------------------------------------------------
template <int N, int KD>
__global__ void gemm_wmma_f32_t(const float* __restrict__ A,
                                const float* __restrict__ W,
                                const float* __restrict__ bias,
                                const float* __restrict__ resid,
                                const float* __restrict__ coordRow,
                                float* __restrict__ C,
                                float scale, int relu, int headMajor)
{
    __shared__ float panel[KD * 16];          // W[:, n0 .. n0+15], row-major
    const int lane = threadIdx.x & 31;
    const int wv   = threadIdx.x >> 5;
    const int tn   = blockIdx.x;
    const int n0   = tn * 16;

    // --- async-stage W panel into LDS (16B per lane-op, ASYNCcnt tracked) ---
    {
        constexpr int chunks = (KD * 16) / 4 / 256;   // b128 ops per thread
        #pragma unroll
        for (int c = 0; c < chunks; ++c) {
            const int elem4 = threadIdx.x + c * 256;  // 4-float group id
            const int k   = elem4 >> 2;
            const int sub = elem4 & 3;
            const float* gsrc = W + (size_t)k * N + n0 + sub * 4;
            unsigned long long ga = (unsigned long long)(uintptr_t)gsrc;
            unsigned ldsOff = (unsigned)(uintptr_t)(panel + k * 16 + sub * 4);
            asm volatile("global_load_async_to_lds_b128 %0, %1, off"
                         :: "v"(ldsOff), "v"(ga) : "memory");
        }
        asm volatile("s_wait_asynccnt 0" ::: "memory");
        __syncthreads();
    }

    const int half = lane >> 4;               // K pair {0,1} vs {2,3}
    const int l16  = lane & 15;
    const int tm   = blockIdx.y * 8 + wv;
    const int m    = tm * 16 + l16;           // A row handled by this lane
    const int n    = n0 + l16;                // C col handled by this lane

    const float* __restrict__ Arow = A + (size_t)m * KD;
    v8f acc = {};

    #pragma unroll 16
    for (int k0 = 0; k0 < KD; k0 += 4) {
        const int ka = k0 + half * 2;
        v2f a = *(const v2f*)(Arow + ka);     // A[m][ka], A[m][ka+1]
        v2f b;
        b.x = panel[ka * 16 + l16];           // B[ka][n]      (LDS)
        b.y = panel[(ka + 1) * 16 + l16];     // B[ka+1][n]    (LDS)
        acc = __builtin_amdgcn_wmma_f32_16x16x4_f32(
            /*neg_a=*/false, a, /*neg_b=*/false, b,
            /*c_mod=*/(short)0, acc, /*reuse_a=*/false, /*reuse_b=*/false);
    }

    const float bn = bias     ? bias[n]     : 0.0f;
    const float cr = coordRow ? coordRow[n] : 0.0f;

    #pragma unroll
    for (int g = 0; g < 8; ++g) {
        const int mrow = tm * 16 + g + half * 8;     // C row for acc[g]
        float v = acc[g];
        if (coordRow) v += cr * ((float)(mrow % TT) * (1.0f / (float)TT));
        v = v * scale + bn;
        if (relu) v = fmaxf(v, 0.0f);
        if (resid) v += resid[(size_t)mrow * N + n];
        size_t off;
        if (headMajor) {
            const int b_ = mrow / TT, t_ = mrow % TT;
            const int h_ = n / EE,   e_ = n % EE;
            off = (((size_t)(b_ * HH + h_)) * TT + t_) * EE + e_;
        } else {
            off = (size_t)mrow * N + n;
        }
        C[off] = v;
    }
}

// ---------------------------------------------------------------------------
// Hyper tail: 512->8 projection, gaussian means/sigmas, index sampling,
// duplicate/causal masking, per-gaussian normalized weights. Wave per row.
// ---------------------------------------------------------------------------
__global__ void hyper_tail(const float* __restrict__ h1,
                           const float* __restrict__ wp2,
                           const float* __restrict__ bp2,
                           int* __restrict__ indbuf,
                           float* __restrict__ wbuf)
{
    const int lane = threadIdx.x & 31;
    const int row  = blockIdx.x * (blockDim.x >> 5) + (threadIdx.x >> 5);
    if (row >= BB * TT) return;
    const int b = row / TT, t = row % TT;

    const float* __restrict__ hr = h1 + (size_t)row * HID;
    float p[8] = {0, 0, 0, 0, 0, 0, 0, 0};
    for (int i = 0; i < HID / 32; ++i) {
        const int k = lane + 32 * i;
        const float hv = hr[k];
        const float* __restrict__ wr = wp2 + (size_t)k * 8;
        #pragma unroll
        for (int j = 0; j < 8; ++j) p[j] += hv * wr[j];
    }
    #pragma unroll
    for (int j = 0; j < 8; ++j) {
        float s = p[j];
        for (int off = 16; off; off >>= 1) s += __shfl_xor(s, off, 32);
        p[j] = s + bp2[j];
    }

    float means[NK], sigmas[NK];
    #pragma unroll
    for (int kk = 0; kk < NK; ++kk) {
        float mu = (float)t - 3.0f * softplusf(p[kk]);
        means[kk]  = fminf(fmaxf(mu, 0.0f), (float)(TT - 1));
        sigmas[kk] = (softplusf(p[NK + kk] + 2.0f) + 0.05f) * 0.1f;
    }

    // lane v owns sample v: kk = v/8, j = v%8 (fm, fm+1, 2 global, 4 local)
    const int kk = lane >> 3, j = lane & 7;
    const int fm = (int)floorf(means[kk]);
    const unsigned key = (((unsigned)(b * TT + t) * NK + (unsigned)kk) << 3) | (unsigned)j;
    int ind;
    if (j == 0)      ind = fm;
    else if (j == 1) ind = min(fm + 1, TT - 1);
    else if (j < 4)  ind = (int)(pcg_hash(key ^ 0xA511E9B3u) & (TT - 1));
    else {
        const int r = (int)(pcg_hash(key ^ 0x3C6EF372u) & 63u);
        ind = min(max(fm + r - 32, 0), TT - 1);
    }

    float pr[NK];
    #pragma unroll
    for (int c = 0; c < NK; ++c) {
        const float d = ((float)ind - means[c]) / sigmas[c];
        pr[c] = __expf(-0.5f * d * d);
    }
    bool dup = false;
    for (int w2 = 0; w2 < 32; ++w2) {
        const int iw = __shfl(ind, w2, 32);
        dup = dup || ((w2 < lane) && (iw == ind));
    }
    if (dup || (ind > t)) {
        #pragma unroll
        for (int c = 0; c < NK; ++c) pr[c] = 0.0f;
    }
    float wsum = 0.0f;
    #pragma unroll
    for (int c = 0; c < NK; ++c) {
        float s = pr[c];
        for (int off = 16; off; off >>= 1) s += __shfl_xor(s, off, 32);
        wsum += pr[c] / s;
    }
    indbuf[(size_t)row * VS + lane] = ind;
    wbuf[(size_t)row * VS + lane]  = wsum;
}

// ---------------------------------------------------------------------------
// Sparse gather attention: wave per (b,h,t); lane = sampled column.
// ---------------------------------------------------------------------------
__global__ void sparse_attn(const float* __restrict__ qb,
                            const float* __restrict__ kb,
                            const float* __restrict__ vb,
                            const int* __restrict__ indbuf,
                            const float* __restrict__ wbuf,
                            float* __restrict__ att)
{
    const int lane = threadIdx.x & 31;
    const int wid  = blockIdx.x * (blockDim.x >> 5) + (threadIdx.x >> 5);
    if (wid >= BB * HH * TT) return;
    const int t  = wid % TT;
    const int bh = wid / TT;               // b*H + h
    const int b  = bh / HH, h = bh % HH;
    const int row = b * TT + t;

    const int   idx = indbuf[(size_t)row * VS + lane];
    const float wgt = wbuf[(size_t)row * VS + lane];

    const float* __restrict__ qr = qb + ((size_t)bh * TT + t) * EE;
    const float* __restrict__ kr = kb + ((size_t)bh * TT + idx) * EE;
    float dot = 0.0f;
    #pragma unroll 8
    for (int e = 0; e < EE; e += 4) {
        const float4 qv = *(const float4*)(qr + e);
        const float4 kv = *(const float4*)(kr + e);
        dot += qv.x * kv.x + qv.y * kv.y + qv.z * kv.z + qv.w * kv.w;
    }
    const float logit = wgt * dot;
    float mx = logit;
    for (int off = 16; off; off >>= 1) mx = fmaxf(mx, __shfl_xor(mx, off, 32));
    const float ex = __expf(logit - mx);
    float s = ex;
    for (int off = 16; off; off >>= 1) s += __shfl_xor(s, off, 32);
    const float prob = ex / s;

    float4 o = {0.0f, 0.0f, 0.0f, 0.0f};
    for (int v2 = 0; v2 < 32; ++v2) {
        const float pv = __shfl(prob, v2, 32);
        const int   iv = __shfl(idx,  v2, 32);
        const float4 vv = *(const float4*)(vb + ((size_t)bh * TT + iv) * EE + lane * 4);
        o.x += pv * vv.x; o.y += pv * vv.y; o.z += pv * vv.z; o.w += pv * vv.w;
    }
    *(float4*)(att + (size_t)row * (HH * EE) + h * EE + lane * 4) = o;
}

// ---------------------------------------------------------------------------
// LayerNorm over E=128, wave per row (4 elements per lane).
// ---------------------------------------------------------------------------
__global__ void layernorm_k(const float* __restrict__ xin,
                            const float* __restrict__ g,
                            const float* __restrict__ be,
                            float* __restrict__ y)
{
    const int lane = threadIdx.x & 31;
    const int row  = blockIdx.x * (blockDim.x >> 5) + (threadIdx.x >> 5);
    if (row >= BB * TT) return;
    const float* __restrict__ xr = xin + (size_t)row * EE;
    const float4 v = *(const float4*)(xr + lane * 4);
    float s  = v.x + v.y + v.z + v.w;
    float s2 = v.x * v.x + v.y * v.y + v.z * v.z + v.w * v.w;
    for (int off = 16; off; off >>= 1) {
        s  += __shfl_xor(s, off, 32);
        s2 += __shfl_xor(s2, off, 32);
    }
    const float mean = s * (1.0f / EE);
    const float var  = s2 * (1.0f / EE) - mean * mean;
    const float r    = rsqrtf(var + 1e-5f);
    const float4 gg = *(const float4*)(g + lane * 4);
    const float4 bb = *(const float4*)(be + lane * 4);
    float4 o;
    o.x = (v.x - mean) * r * gg.x + bb.x;
    o.y = (v.y - mean) * r * gg.y + bb.y;
    o.z = (v.z - mean) * r * gg.z + bb.z;
    o.w = (v.w - mean) * r * gg.w + bb.w;
    *(float4*)(y + (size_t)row * EE + lane * 4) = o;
}

// ---------------------------------------------------------------------------
extern "C" void kernel_launch(void* const* d_in, const int* in_sizes, int n_in,
                              void* d_out, int out_size, void* d_ws, size_t ws_size,
                              hipStream_t stream)
{
    (void)in_sizes; (void)n_in; (void)out_size; (void)ws_size;
    const float* x   = (const float*)d_in[0];
    const float* wq  = (const float*)d_in[1];
    const float* wk  = (const float*)d_in[2];
    const float* wv  = (const float*)d_in[3];
    const float* wu  = (const float*)d_in[4];
    const float* bu  = (const float*)d_in[5];
    const float* wp1 = (const float*)d_in[6];
    const float* bp1 = (const float*)d_in[7];
    const float* wp2 = (const float*)d_in[8];
    const float* bp2 = (const float*)d_in[9];
    const float* g1  = (const float*)d_in[10];
    const float* be1 = (const float*)d_in[11];
    const float* g2  = (const float*)d_in[12];
    const float* be2 = (const float*)d_in[13];
    const float* wf1 = (const float*)d_in[14];
    const float* bf1 = (const float*)d_in[15];
    const float* wf2 = (const float*)d_in[16];
    const float* bf2 = (const float*)d_in[17];

    const size_t MR = (size_t)BB * TT;       // 2048 rows
    float* ws  = (float*)d_ws;
    float* qb  = ws;                         // (b,h,t,e) : 2048*512
    float* kb  = qb  + MR * HID;
    float* vb  = kb  + MR * HID;
    float* h1  = vb  + MR * HID;             // hyper hidden (2048,512)
    float* att = h1  + MR * HID;             // (2048,512)
    float* h2  = att + MR * HID;             // ffn hidden (2048,512)
    float* s1  = h2  + MR * HID;             // pre-LN1 (2048,128)
    float* x1  = s1  + MR * EE;
    float* s2  = x1  + MR * EE;
    int*   ib  = (int*)(s2 + MR * EE);       // (2048,32) indices
    float* wb  = (float*)(ib + MR * VS);     // (2048,32) weights

    const float scale = 0.29730177875068026f;   // 128^-0.25
    const dim3 blk(256);                        // 8 waves per block
    const dim3 gBig(512 / 16, 2048 / 16 / 8);   // N-tiles x M-tile-groups
    const dim3 gSmall(128 / 16, 2048 / 16 / 8);
    const dim3 gRow(2048 / 8);
    const dim3 gAttn(BB * HH * TT / 8);

    // Q, K, V projections (head-major output for the gather stage)
    gemm_wmma_f32_t<512, 128><<<gBig, blk, 0, stream>>>(x, wq, nullptr, nullptr, nullptr, qb,
                                                        scale, 0, 1);
    gemm_wmma_f32_t<512, 128><<<gBig, blk, 0, stream>>>(x, wk, nullptr, nullptr, nullptr, kb,
                                                        scale, 0, 1);
    gemm_wmma_f32_t<512, 128><<<gBig, blk, 0, stream>>>(x, wv, nullptr, nullptr, nullptr, vb,
                                                        1.0f, 0, 1);
    // Hyper net layer 1: relu(x@wp1[:128] + coord*wp1[128] + bp1)
    gemm_wmma_f32_t<512, 128><<<gBig, blk, 0, stream>>>(x, wp1, bp1, nullptr, wp1 + 128 * HID,
                                                        h1, 1.0f, 1, 0);
    // Hyper tail: params -> means/sigmas -> sampled indices + weights
    hyper_tail<<<gRow, blk, 0, stream>>>(h1, wp2, bp2, ib, wb);
    // Sparse gather attention
    sparse_attn<<<gAttn, blk, 0, stream>>>(qb, kb, vb, ib, wb, att);
    // Output projection + bias + residual -> s1 ; LN -> x1
    gemm_wmma_f32_t<128, 512><<<gSmall, blk, 0, stream>>>(att, wu, bu, x, nullptr, s1,
                                                          1.0f, 0, 0);
    layernorm_k<<<gRow, blk, 0, stream>>>(s1, g1, be1, x1);
    // FFN
    gemm_wmma_f32_t<512, 128><<<gBig, blk, 0, stream>>>(x1, wf1, bf1, nullptr, nullptr, h2,
                                                        1.0f, 1, 0);
    gemm_wmma_f32_t<128, 512><<<gSmall, blk, 0, stream>>>(h2, wf2, bf2, x1, nullptr, s2,
                                                          1.0f, 0, 0);
    layernorm_k<<<gRow, blk, 0, stream>>>(s2, g2, be2, (float*)d_out);
}